// ResampleDense_24223615549696
// MI455X (gfx1250) — compile-verified
//
#include <hip/hip_runtime.h>
#include <hip/hip_bf16.h>

#define NPTS   32768
#define NKER   4
#define GRD    16
#define CIN    64
#define COUT   64
#define NCELLS (NKER * GRD * GRD * GRD)   /* 16384 */
#define NENT   (NPTS * 8)                 /* 262144 */

typedef __attribute__((ext_vector_type(2))) float v2f;
typedef __attribute__((ext_vector_type(8))) float v8f;

// ---------------------------------------------------------------- pass 0
__global__ void rd_zero(int* __restrict__ counts, int* __restrict__ cursor) {
    int i = blockIdx.x * blockDim.x + threadIdx.x;
    if (i < NCELLS) { counts[i] = 0; cursor[i] = 0; }
}

// ---------------------------------------------------------------- pass 1
// Per point: 8 corner cell ids + trilinear weights, histogram of cells.
__global__ void rd_prep(const int* __restrict__ pidx, const float* __restrict__ pos,
                        int* __restrict__ cells, float* __restrict__ wts,
                        int* __restrict__ counts) {
    int n = blockIdx.x * blockDim.x + threadIdx.x;
    if (n >= NPTS) return;
    int p = pidx[n];
    float l0 = pos[n * 3 + 0] * (float)GRD - 0.5f;
    float l1 = pos[n * 3 + 1] * (float)GRD - 0.5f;
    float l2 = pos[n * 3 + 2] * (float)GRD - 0.5f;
    float f0 = floorf(l0), f1 = floorf(l1), f2 = floorf(l2);
    float c0 = l0 - f0, c1 = l1 - f1, c2 = l2 - f2;
    int i0 = (int)f0, i1 = (int)f1, i2 = (int)f2;
#pragma unroll
    for (int c = 0; c < 8; ++c) {
        // CORNERS[c] = [ (c>>2)&1, (c>>1)&1, c&1 ] over pos components 0,1,2
        int b0 = (c >> 2) & 1, b1 = (c >> 1) & 1, b2 = c & 1;
        int x0 = min(max(i0 + b0, 0), GRD - 1);
        int x1 = min(max(i1 + b1, 0), GRD - 1);
        int x2 = min(max(i2 + b2, 0), GRD - 1);
        float w = (b0 ? c0 : 1.f - c0) * (b1 ? c1 : 1.f - c1) * (b2 ? c2 : 1.f - c2);
        // kernels indexed [p, idx2, idx1, idx0]
        int cell = ((p * GRD + x2) * GRD + x1) * GRD + x0;
        cells[n * 8 + c] = cell;
        wts[n * 8 + c]   = w;
        atomicAdd(&counts[cell], 1);
    }
}

// ---------------------------------------------------------------- pass 2
// out[n][f] = sum_c w_c * bias_cell[f]   (initializes d_out deterministically)
__global__ void rd_bias(const float* __restrict__ biases, const int* __restrict__ cells,
                        const float* __restrict__ wts, float* __restrict__ out) {
    int t = blockIdx.x * blockDim.x + threadIdx.x;
    if (t >= NPTS * COUT) return;
    int n = t >> 6, f = t & (COUT - 1);
    float s = 0.f;
#pragma unroll
    for (int c = 0; c < 8; ++c) {
        int e = n * 8 + c;
        s += wts[e] * biases[(size_t)cells[e] * COUT + f];
    }
    out[t] = s;
}

// ---------------------------------------------------------------- pass 3
// Exclusive prefix sum over 16384 counts, single block of 256 threads.
__global__ void rd_scan(const int* __restrict__ counts, int* __restrict__ offsets) {
    __shared__ int partial[256];
    __shared__ int excl[256];
    int t = threadIdx.x;
    int base = t * (NCELLS / 256);
    int s = 0;
    for (int i = 0; i < NCELLS / 256; ++i) s += counts[base + i];
    partial[t] = s;
    __syncthreads();
    if (t == 0) {
        int run = 0;
        for (int i = 0; i < 256; ++i) { excl[i] = run; run += partial[i]; }
    }
    __syncthreads();
    int run = excl[t];
    for (int i = 0; i < NCELLS / 256; ++i) {
        offsets[base + i] = run;
        run += counts[base + i];
    }
}

// ---------------------------------------------------------------- pass 4
// Scatter entry ids (n<<3 | corner) into cell-sorted order.
__global__ void rd_scatter(const int* __restrict__ cells, const int* __restrict__ offsets,
                           int* __restrict__ cursor, int* __restrict__ entries) {
    int e = blockIdx.x * blockDim.x + threadIdx.x;
    if (e >= NENT) return;
    int cell = cells[e];
    int slot = offsets[cell] + atomicAdd(&cursor[cell], 1);
    entries[slot] = e;
}

// ---------------------------------------------------------------- pass 5
// Per cell: D(P x 64) = A(P x 64) @ Kcell(64 x 64) via v_wmma_f32_16x16x4_f32.
// Block = 128 threads = 4 waves; wave j owns output columns 16j..16j+15.
__global__ __launch_bounds__(128) void rd_gemm(const float* __restrict__ xs,
                                               const float* __restrict__ kernels,
                                               const int* __restrict__ entries,
                                               const int* __restrict__ offsets,
                                               const int* __restrict__ counts,
                                               const float* __restrict__ wts,
                                               float* __restrict__ out) {
    int cell  = blockIdx.x;
    int cnt   = counts[cell];
    if (cnt == 0) return;
    int start = offsets[cell];

    int wave = threadIdx.x >> 5;        // 0..3 : column tile
    int lane = threadIdx.x & 31;
    int nloc = lane & 15;               // M index for A rows, N index for B cols
    int h    = lane >> 4;               // K-half selector (ISA A/B layout)
    int col  = wave * 16 + nloc;

    const float* __restrict__ B = kernels + (size_t)cell * (CIN * COUT);

    for (int chunk = 0; chunk < cnt; chunk += 16) {
        int rows = min(16, cnt - chunk);

        // A row for this lane (clamped for partial tiles; garbage rows discarded on store)
        int eA = entries[start + chunk + min(nloc, rows - 1)];
        const float* __restrict__ arow = xs + (size_t)(eA >> 3) * CIN;

        v8f acc = {};
#pragma unroll
        for (int i = 0; i < CIN / 4; ++i) {
            int k = 4 * i + 2 * h;
            // A 16x4 fragment: lanes 0-15 hold K=k0,k0+1 ; lanes 16-31 hold K=k0+2,k0+3
            v2f a = *(const v2f*)(arow + k);
            // B 4x16 fragment: same K striping, N = lane%16
            v2f b;
            b.x = B[(size_t)k * COUT + col];
            b.y = B[(size_t)(k + 1) * COUT + col];
            acc = __builtin_amdgcn_wmma_f32_16x16x4_f32(
                /*neg_a=*/false, a, /*neg_b=*/false, b,
                /*c_mod=*/(short)0, acc, /*reuse_a=*/false, /*reuse_b=*/false);
        }

        // D layout: VGPR r -> M=r (lanes 0-15) / M=8+r (lanes 16-31); N = lane%16
#pragma unroll
        for (int r = 0; r < 8; ++r) {
            int mm = h * 8 + r;
            if (mm < rows) {
                int e = entries[start + chunk + mm];
                float w = wts[e];
                atomicAdd(&out[(size_t)(e >> 3) * COUT + col], w * acc[r]);
            }
        }
    }
}

// ----------------------------------------------------------------
extern "C" void kernel_launch(void* const* d_in, const int* in_sizes, int n_in,
                              void* d_out, int out_size, void* d_ws, size_t ws_size,
                              hipStream_t stream) {
    const int*   pidx    = (const int*)  d_in[0];   // (N,1) int32
    const float* pos     = (const float*)d_in[1];   // (N,3)
    const float* xs      = (const float*)d_in[2];   // (N,64)
    const float* kernels = (const float*)d_in[3];   // (4,16,16,16,64,64)
    const float* biases  = (const float*)d_in[4];   // (4,16,16,16,64)
    float*       out     = (float*)d_out;           // (N,64)

    char* w = (char*)d_ws;
    int*   cells   = (int*)  w; w += (size_t)NENT   * sizeof(int);
    float* wts     = (float*)w; w += (size_t)NENT   * sizeof(float);
    int*   counts  = (int*)  w; w += (size_t)NCELLS * sizeof(int);
    int*   offsets = (int*)  w; w += (size_t)NCELLS * sizeof(int);
    int*   cursor  = (int*)  w; w += (size_t)NCELLS * sizeof(int);
    int*   entries = (int*)  w; w += (size_t)NENT   * sizeof(int);
    (void)ws_size; (void)in_sizes; (void)n_in; (void)out_size;

    rd_zero   <<<(NCELLS + 255) / 256, 256, 0, stream>>>(counts, cursor);
    rd_prep   <<<(NPTS   + 255) / 256, 256, 0, stream>>>(pidx, pos, cells, wts, counts);
    rd_bias   <<<(NPTS * COUT + 255) / 256, 256, 0, stream>>>(biases, cells, wts, out);
    rd_scan   <<<1, 256, 0, stream>>>(counts, offsets);
    rd_scatter<<<(NENT + 255) / 256, 256, 0, stream>>>(cells, offsets, cursor, entries);
    rd_gemm   <<<NCELLS, 128, 0, stream>>>(xs, kernels, entries, offsets, counts, wts, out);
}